// CrossKDLoss_69483980915432
// MI455X (gfx1250) — compile-verified
//
#include <hip/hip_runtime.h>
#include <hip/hip_bf16.h>
#include <math.h>
#include <stdint.h>

// ---------------------------------------------------------------------------
// CrossKD distillation loss for MI455X (gfx1250, wave32).
//  - prep_kernel    : xywh->xyxy + areas, teacher conf mask (with argmax fallback)
//  - greedy_kernel  : serial greedy IoU matching, 1 workgroup per (level,image),
//                     teacher data staged in LDS via global_load_async_to_lds_*
//  - loss_kernel    : per 16-row tile: softmax/log-softmax (80 classes), KL cross
//                     term via chained V_WMMA_F32_16X16X4_F32 (diag of 16x16 GEMM),
//                     box L1 and conf MSE; deterministic per-tile partials
//  - finalize_kernel: fixed-order reduction -> scalar
// ---------------------------------------------------------------------------

#define KALPHA 0.6f
#define KBETA  0.3f
#define KTEMP  4.0f
#define KIOU   0.5f
#define KCONF  0.5f

typedef float v2f __attribute__((ext_vector_type(2)));
typedef float v8f __attribute__((ext_vector_type(8)));

// workspace layout (units of float); 8 units: u<4 -> level0 (N=2048), u>=4 -> level1 (N=1024)
#define UNIT_STRIDE 26624
#define OFF_SBOX   0        // float4 x N  (student xyxy)
#define OFF_TBOX   8192     // float4 x N  (teacher xyxy)
#define OFF_SAREA  16384    // float  x N
#define OFF_TAREA  18432    // float  x N
#define OFF_TMASK  20480    // float  x N
#define OFF_W      22528    // float  x N  (match weight 0/1)
#define OFF_MIOU   24576    // float  x N  (matched iou or 0)
#define OFF_TIDX   212992   // int    x 2048 per unit (8 units)
#define OFF_M      229376   // float  x 8  (match count per unit)
#define OFF_PART   229384   // float  x 768*3 (per-tile cls/box/conf partials)
#define NTILES     768      // 4*128 + 4*64

__device__ __forceinline__ int unitN(int u) { return u < 4 ? 2048 : 1024; }

// ---------------------------------------------------------------------------
__global__ __launch_bounds__(1024)
void prep_kernel(const float* __restrict__ s0, const float* __restrict__ t0,
                 const float* __restrict__ s1, const float* __restrict__ t1,
                 float* __restrict__ ws) {
    int u = blockIdx.x;
    int N = unitN(u);
    const float* sg = (u < 4 ? s0 : s1) + (size_t)(u & 3) * N * 85;
    const float* tg = (u < 4 ? t0 : t1) + (size_t)(u & 3) * N * 85;
    float* wsU = ws + (size_t)u * UNIT_STRIDE;
    float4* sbox = (float4*)(wsU + OFF_SBOX);
    float4* tbox = (float4*)(wsU + OFF_TBOX);
    float* sarea = wsU + OFF_SAREA;
    float* tarea = wsU + OFF_TAREA;
    float* tmask = wsU + OFF_TMASK;

    int tid = threadIdx.x;
    float bc = -INFINITY; int bi = 0x7fffffff;  // argmax(conf), lowest-index ties
    for (int r = tid; r < N; r += 1024) {
        float xc = sg[r*85+0], yc = sg[r*85+1], w = sg[r*85+2], h = sg[r*85+3];
        float x1 = xc - 0.5f*w, y1 = yc - 0.5f*h, x2 = xc + 0.5f*w, y2 = yc + 0.5f*h;
        sbox[r] = make_float4(x1, y1, x2, y2);
        sarea[r] = (x2 - x1) * (y2 - y1);            // areas from xyxy, like reference
        xc = tg[r*85+0]; yc = tg[r*85+1]; w = tg[r*85+2]; h = tg[r*85+3];
        x1 = xc - 0.5f*w; y1 = yc - 0.5f*h; x2 = xc + 0.5f*w; y2 = yc + 0.5f*h;
        tbox[r] = make_float4(x1, y1, x2, y2);
        tarea[r] = (x2 - x1) * (y2 - y1);
        float conf = tg[r*85+4];
        if (conf > bc || (conf == bc && r < bi)) { bc = conf; bi = r; }
    }
    __shared__ float rV[32]; __shared__ int rI[32];
    __shared__ float smc;    __shared__ int smi;
    for (int off = 16; off; off >>= 1) {
        float ov = __shfl_down(bc, off, 32); int oi = __shfl_down(bi, off, 32);
        if (ov > bc || (ov == bc && oi < bi)) { bc = ov; bi = oi; }
    }
    int wv = tid >> 5;
    if ((tid & 31) == 0) { rV[wv] = bc; rI[wv] = bi; }
    __syncthreads();
    if (wv == 0) {
        bc = rV[tid]; bi = rI[tid];   // 1024 threads -> exactly 32 waves
        for (int off = 16; off; off >>= 1) {
            float ov = __shfl_down(bc, off, 32); int oi = __shfl_down(bi, off, 32);
            if (ov > bc || (ov == bc && oi < bi)) { bc = ov; bi = oi; }
        }
        if (tid == 0) { smc = bc; smi = bi; }
    }
    __syncthreads();
    bool anym = smc > KCONF;   // any(conf>thr) <=> max(conf)>thr
    for (int r = tid; r < N; r += 1024) {
        float conf = tg[r*85+4];
        tmask[r] = anym ? (conf > KCONF ? 1.f : 0.f) : (r == smi ? 1.f : 0.f);
    }
}

// ---------------------------------------------------------------------------
__global__ __launch_bounds__(1024)
void greedy_kernel(float* __restrict__ ws, int* __restrict__ tidxAll) {
    int u = blockIdx.x;
    int N = unitN(u);
    float* wsU = ws + (size_t)u * UNIT_STRIDE;
    const float4* tbox_g  = (const float4*)(wsU + OFF_TBOX);
    const float*  tarea_g = wsU + OFF_TAREA;
    const float*  tmask_g = wsU + OFF_TMASK;
    const float4* sbox_g  = (const float4*)(wsU + OFF_SBOX);
    const float*  sarea_g = wsU + OFF_SAREA;
    float* w_out    = wsU + OFF_W;
    float* miou_out = wsU + OFF_MIOU;
    int*   tidx_out = tidxAll + u * 2048;

    __shared__ float4 tb[2048];     // 32 KB
    __shared__ float  ta[2048];     //  8 KB
    __shared__ float  tm[2048];     //  8 KB
    __shared__ int    used[2048];   //  8 KB
    __shared__ float  rV[32]; __shared__ int rI[32];

    int tid = threadIdx.x;
    // Stage teacher boxes/areas/mask into LDS via CDNA5 async-to-LDS path.
    for (int j = tid; j < N; j += 1024) {
        unsigned la  = (unsigned)(uintptr_t)(&tb[j]);
        unsigned la2 = (unsigned)(uintptr_t)(&ta[j]);
        unsigned la3 = (unsigned)(uintptr_t)(&tm[j]);
        unsigned long long ga  = (unsigned long long)(uintptr_t)(tbox_g + j);
        unsigned long long ga2 = (unsigned long long)(uintptr_t)(tarea_g + j);
        unsigned long long ga3 = (unsigned long long)(uintptr_t)(tmask_g + j);
        asm volatile("global_load_async_to_lds_b128 %0, %1, off" :: "v"(la),  "v"(ga)  : "memory");
        asm volatile("global_load_async_to_lds_b32  %0, %1, off" :: "v"(la2), "v"(ga2) : "memory");
        asm volatile("global_load_async_to_lds_b32  %0, %1, off" :: "v"(la3), "v"(ga3) : "memory");
        used[j] = 0;
    }
    asm volatile("s_wait_asynccnt 0" ::: "memory");
    __syncthreads();

    float Mcount = 0.f;
    for (int i = 0; i < N; ++i) {
        float4 sb = sbox_g[i];
        float  sA = sarea_g[i];
        float bv = -INFINITY; int bj = 0x7fffffff;
        for (int j = tid; j < N; j += 1024) {
            float v = -INFINITY;
            if (tm[j] != 0.f && used[j] == 0) {
                float4 t = tb[j];
                float tlx = fmaxf(sb.x, t.x), tly = fmaxf(sb.y, t.y);
                float brx = fminf(sb.z, t.z), bry = fminf(sb.w, t.w);
                float iw = fmaxf(brx - tlx, 0.f), ih = fmaxf(bry - tly, 0.f);
                float inter = iw * ih;
                v = inter / (sA + ta[j] - inter + 1e-7f);
            }
            if (v > bv || (v == bv && j < bj)) { bv = v; bj = j; }
        }
        for (int off = 16; off; off >>= 1) {
            float ov = __shfl_down(bv, off, 32); int oj = __shfl_down(bj, off, 32);
            if (ov > bv || (ov == bv && oj < bj)) { bv = ov; bj = oj; }
        }
        int wv = tid >> 5;
        if ((tid & 31) == 0) { rV[wv] = bv; rI[wv] = bj; }
        __syncthreads();
        if (wv == 0) {
            bv = rV[tid]; bj = rI[tid];
            for (int off = 16; off; off >>= 1) {
                float ov = __shfl_down(bv, off, 32); int oj = __shfl_down(bj, off, 32);
                if (ov > bv || (ov == bv && oj < bj)) { bv = ov; bj = oj; }
            }
            if (tid == 0) {
                bool matched = bv > KIOU;
                if (matched) { used[bj] = 1; Mcount += 1.f; }
                w_out[i]    = matched ? 1.f : 0.f;
                miou_out[i] = matched ? bv : 0.f;
                tidx_out[i] = bj;
            }
        }
        __syncthreads();
    }
    if (tid == 0) ws[OFF_M + u] = Mcount;
}

// ---------------------------------------------------------------------------
// 4 waves/block, each wave owns one 16-row tile. KL cross term via WMMA f32.
__global__ __launch_bounds__(128)
void loss_kernel(const float* __restrict__ s0, const float* __restrict__ t0,
                 const float* __restrict__ s1, const float* __restrict__ t1,
                 float* __restrict__ ws, const int* __restrict__ tidxAll) {
    __shared__ float Asl[4][16][80];   // s_log tiles (A: 16xK)
    __shared__ float Bts[4][16][80];   // t_soft tiles (B: KxN, stored N-major)
    __shared__ float Tl[4][16];        // sum t*log t per row
    __shared__ float Cd[4][8][32];     // WMMA D dump for diagonal extraction

    int wvid = threadIdx.x >> 5;
    int lane = threadIdx.x & 31;
    int gt = blockIdx.x * 4 + wvid;               // 0..767
    int u, tile;
    if (gt < 512) { u = gt >> 7;              tile = gt & 127; }
    else          { u = 4 + ((gt - 512) >> 6); tile = (gt - 512) & 63; }
    int N = unitN(u);
    const float* sg = (u < 4 ? s0 : s1) + (size_t)(u & 3) * N * 85;
    const float* tg = (u < 4 ? t0 : t1) + (size_t)(u & 3) * N * 85;
    float* wsU = ws + (size_t)u * UNIT_STRIDE;
    const float* w_ws    = wsU + OFF_W;
    const float* miou_ws = wsU + OFF_MIOU;
    const int*   tidx_ws = tidxAll + u * 2048;

    int r = lane & 15;
    int row = tile * 16 + r;
    bool teacher = lane >= 16;                    // lanes 0-15: student, 16-31: teacher
    int tj = tidx_ws[row];
    const float invT = 1.0f / KTEMP;
    const float* src = teacher ? (tg + (size_t)tj * 85 + 5) : (sg + (size_t)row * 85 + 5);

    float m = -INFINITY;
    for (int c = 0; c < 80; ++c) m = fmaxf(m, src[c] * invT);
    float sum = 0.f;
    for (int c = 0; c < 80; ++c) sum += expf(src[c] * invT - m);
    float logden = logf(sum);
    if (teacher) {
        float tl = 0.f;
        for (int c = 0; c < 80; ++c) {
            float lx = src[c] * invT - m - logden;
            float p = expf(lx);
            Bts[wvid][r][c] = p;
            tl += p * lx;
        }
        Tl[wvid][r] = tl;
    } else {
        for (int c = 0; c < 80; ++c) Asl[wvid][r][c] = src[c] * invT - m - logden;
    }
    __syncthreads();

    // D = A(16x80) x B(80x16), accumulate 20 chunks of K=4.
    // A frag layout (32-bit 16x4): lanes 0-15 -> M=lane, K=k0/k0+1; lanes 16-31 -> K=k0+2/k0+3.
    // B frag mirrors it (N=lane over rows K); both index their LDS tile identically.
    v8f acc = {0.f,0.f,0.f,0.f,0.f,0.f,0.f,0.f};
    int koff = teacher ? 2 : 0;
    for (int cc = 0; cc < 20; ++cc) {
        int k0 = cc * 4 + koff;
        v2f a; a[0] = Asl[wvid][r][k0]; a[1] = Asl[wvid][r][k0 + 1];
        v2f b; b[0] = Bts[wvid][r][k0]; b[1] = Bts[wvid][r][k0 + 1];
        acc = __builtin_amdgcn_wmma_f32_16x16x4_f32(false, a, false, b,
                                                    (short)0, acc, false, false);
    }
#pragma unroll
    for (int k = 0; k < 8; ++k) Cd[wvid][k][lane] = acc[k];
    __syncthreads();

    float cls = 0.f, box = 0.f, cf = 0.f;
    if (!teacher) {
        int i = lane;
        // diag(i,i): i<8 -> VGPR i lane i ; i>=8 -> VGPR i-8 lane i+16
        float cross = (i < 8) ? Cd[wvid][i][i] : Cd[wvid][i - 8][i + 16];
        float wgt  = w_ws[row];
        float miou = miou_ws[row];
        float kl = Tl[wvid][i] - cross;
        cls = kl * wgt;
        float bsum = 0.f;
        for (int d = 0; d < 4; ++d)
            bsum += fabsf(sg[(size_t)row * 85 + d] - tg[(size_t)tj * 85 + d]);
        box = bsum * miou * wgt;
        float dcf = sg[(size_t)row * 85 + 4] - tg[(size_t)tj * 85 + 4] * miou;
        cf = dcf * dcf * wgt;
    }
    for (int off = 16; off; off >>= 1) {
        cls += __shfl_down(cls, off, 32);
        box += __shfl_down(box, off, 32);
        cf  += __shfl_down(cf,  off, 32);
    }
    if (lane == 0) {
        float Msafe = fmaxf(ws[OFF_M + u], 1.f);
        float* part = ws + OFF_PART + (size_t)gt * 3;
        part[0] = cls * (KTEMP * KTEMP) / Msafe;
        part[1] = box / (4.f * Msafe);
        part[2] = cf / Msafe;
    }
}

// ---------------------------------------------------------------------------
__global__ void finalize_kernel(const float* __restrict__ ws, float* __restrict__ out) {
    float nm = 0.f;
    for (int u = 0; u < 8; ++u) nm += ws[OFF_M + u];
    float nms = fmaxf(nm, 1.f);
    float cls = 0.f, box = 0.f, cf = 0.f;
    for (int t = 0; t < NTILES; ++t) {            // fixed order -> deterministic
        cls += ws[OFF_PART + t * 3 + 0];
        box += ws[OFF_PART + t * 3 + 1];
        cf  += ws[OFF_PART + t * 3 + 2];
    }
    out[0] = (KALPHA * cls + KBETA * box + (1.f - KALPHA - KBETA) * cf) / nms;
}

// ---------------------------------------------------------------------------
extern "C" void kernel_launch(void* const* d_in, const int* in_sizes, int n_in,
                              void* d_out, int out_size, void* d_ws, size_t ws_size,
                              hipStream_t stream) {
    (void)in_sizes; (void)n_in; (void)out_size; (void)ws_size;
    const float* s0 = (const float*)d_in[0];
    const float* t0 = (const float*)d_in[1];
    const float* s1 = (const float*)d_in[2];
    const float* t1 = (const float*)d_in[3];
    float* ws = (float*)d_ws;
    int* tidxAll = (int*)(ws + OFF_TIDX);

    hipLaunchKernelGGL(prep_kernel,   dim3(8),   dim3(1024), 0, stream, s0, t0, s1, t1, ws);
    hipLaunchKernelGGL(greedy_kernel, dim3(8),   dim3(1024), 0, stream, ws, tidxAll);
    hipLaunchKernelGGL(loss_kernel,   dim3(192), dim3(128),  0, stream, s0, t0, s1, t1, ws, tidxAll);
    hipLaunchKernelGGL(finalize_kernel, dim3(1), dim3(1),    0, stream, ws, (float*)d_out);
}